// AttrMessagePassing_6098853560479
// MI455X (gfx1250) — compile-verified
//
#include <hip/hip_runtime.h>
#include <math.h>

// ---------------------------------------------------------------------------
// Equivariant GNN message passing for MI455X (gfx1250, wave32, WMMA).
// All GEMMs use V_WMMA_F32_16X16X4_F32 (f32-exact). Edge MLP (the dominant
// 10 GFLOP/layer GEMM) is fused into the edge kernel via LDS so the (E,192)
// intermediate never hits HBM. Segment-sum uses hardware f32 global atomics
// into accumulators (77 MB) that stay resident in the 192MB L2.
// ---------------------------------------------------------------------------

#define N_NODES 50000
#define N_EDGES 400000

typedef __attribute__((ext_vector_type(2))) float v2f;
typedef __attribute__((ext_vector_type(8))) float v8f;

__device__ __forceinline__ v8f wmma4(v2f a, v2f b, v8f c) {
  // D = A(16x4) * B(4x16) + C(16x16), f32. 8-arg VOP3P form.
  return __builtin_amdgcn_wmma_f32_16x16x4_f32(
      /*neg_a=*/false, a, /*neg_b=*/false, b,
      /*c_mod=*/(short)0, c, /*reuse_a=*/false, /*reuse_b=*/false);
}

// One wave computes a 16x16 C tile: C = A[m0:m0+16, 0:K] @ B[0:K, n0:n0+16].
// ESA = element stride within an A row (1 = contiguous -> single b64 load,
// 3 = the (N,V,3) vector layouts). B is row-major weights, element stride 1.
// VGPR layouts per CDNA5 ISA 7.12.2:
//   A 16x4 : lanes 0-15 hold K={k,k+1}, lanes 16-31 hold K={k+2,k+3}, M=lane&15
//   B 4x16 : lanes 0-15 hold K={k,k+1}, lanes 16-31 hold K={k+2,k+3}, N=lane&15
//   C 16x16: vgpr r -> M = r + 8*(lane>=16), N = lane&15
template <int ESA>
__device__ __forceinline__ v8f gemm_tile(const float* __restrict__ A, int ldA,
                                         const float* __restrict__ B, int ldB,
                                         int m0, int n0, int K, int lane) {
  v8f acc = {};
  const int mr = m0 + (lane & 15);
  const int nc = n0 + (lane & 15);
  const int kh = (lane >> 4) << 1;  // 0 or 2
  const float* __restrict__ arow = A + (size_t)mr * ldA;
  for (int k = 0; k < K; k += 4) {
    const int kb = k + kh;
    v2f a;
    if constexpr (ESA == 1) {
      a = *(const v2f*)(arow + kb);  // 8B-aligned everywhere we use ESA==1
    } else {
      a.x = arow[(size_t)kb * ESA];
      a.y = arow[(size_t)(kb + 1) * ESA];
    }
    v2f b;
    b.x = B[(size_t)kb * ldB + nc];
    b.y = B[(size_t)(kb + 1) * ldB + nc];
    acc = wmma4(a, b, acc);
  }
  return acc;
}

// Store a 16x16 tile: C[m][n] = (ADD ? add[m][n] : 0) + acc*scale*(RS ? rs[m] : 1)
// RS/ADD are compile-time so the epilogue is branch-free.
template <bool RS, bool ADD>
__device__ __forceinline__ void store_tile(v8f acc, float* C, int ldC, int esC,
                                           int m0, int n0, int lane, float scale,
                                           const float* rowScale,
                                           const float* addB, int ldAdd, int esAdd) {
  const int nc = n0 + (lane & 15);
  const int mh = (lane >> 4) << 3;
#pragma unroll
  for (int r = 0; r < 8; ++r) {
    const int m = m0 + r + mh;
    float v = acc[r] * scale;
    if constexpr (RS) v *= rowScale[m];
    if constexpr (ADD) v += addB[(size_t)m * ldAdd + (size_t)nc * esAdd];
    C[(size_t)m * ldC + (size_t)nc * esC] = v;
  }
}

// ---------------------------------------------------------------------------
// Node-side linear transforms:
//   hs   = xs @ Wl1_s * attr / sqrt(64)          (N,64)
//   sc_s = xs @ Wsc_s * attr / sqrt(64)          (N,o_sc)
//   hv_i = xv_i @ Wl1_v * attr / sqrt(32)        (N,32,3)
//   sc_v_i = xv_i @ Wsc_v * attr / sqrt(32)      (N,32,3)
// One wave per 16-node tile, 4 waves per block.
// ---------------------------------------------------------------------------
__global__ __launch_bounds__(128) void node_linear_kernel(
    const float* __restrict__ xs, const float* __restrict__ xv,
    const float* __restrict__ attr,
    const float* __restrict__ Wsc_s, const float* __restrict__ Wsc_v,
    const float* __restrict__ Wl1_s, const float* __restrict__ Wl1_v,
    float* hs, float* hv, float* sc_s, float* sc_v, int o_sc, int nNodes) {
  const int lane = threadIdx.x & 31;
  const int m0 = blockIdx.x * 64 + (threadIdx.x >> 5) * 16;
  if (m0 >= nNodes) return;
  const float ss = 0.125f;                // 1/sqrt(64)
  const float sv = 0.17677669529663687f;  // 1/sqrt(32)
  for (int nt = 0; nt < 64; nt += 16) {
    v8f acc = gemm_tile<1>(xs, 64, Wl1_s, 64, m0, nt, 64, lane);
    store_tile<true, false>(acc, hs, 64, 1, m0, nt, lane, ss, attr, nullptr, 0, 0);
  }
  for (int nt = 0; nt < o_sc; nt += 16) {
    v8f acc = gemm_tile<1>(xs, 64, Wsc_s, o_sc, m0, nt, 64, lane);
    store_tile<true, false>(acc, sc_s, o_sc, 1, m0, nt, lane, ss, attr, nullptr, 0, 0);
  }
#pragma unroll
  for (int i = 0; i < 3; ++i) {
    for (int nt = 0; nt < 32; nt += 16) {
      v8f acc = gemm_tile<3>(xv + i, 96, Wl1_v, 32, m0, nt, 32, lane);
      store_tile<true, false>(acc, hv + i, 96, 3, m0, nt, lane, sv, attr, nullptr, 0, 0);
      acc = gemm_tile<3>(xv + i, 96, Wsc_v, 32, m0, nt, 32, lane);
      store_tile<true, false>(acc, sc_v + i, 96, 3, m0, nt, lane, sv, attr, nullptr, 0, 0);
    }
  }
}

// ---------------------------------------------------------------------------
// Fused edge kernel: 32 edges per block of 256 threads (8 waves).
//   1. esc tile (32x10) -> LDS
//   2. h = silu(esc @ Wfc1 / sqrt(10)) -> LDS (VALU, K=10 too small for WMMA)
//   3. w = h @ Wfc2 / sqrt(64) -> LDS via WMMA (32x192, 24 tiles across 8 waves)
//      Wfc2 (48KB) is shared by all blocks -> stays hot in WGP$/L2.
//   4. gather hs/hv at src, build messages, atomic-scatter to Ms/Mv/deg
// ---------------------------------------------------------------------------
__global__ __launch_bounds__(256) void edge_kernel(
    const float* __restrict__ esc, const float* __restrict__ Wfc1,
    const float* __restrict__ Wfc2, const float* __restrict__ hs,
    const float* __restrict__ hv, const int* __restrict__ src,
    const int* __restrict__ dst, const float* __restrict__ sh,
    float* Ms, float* Mv, float* deg) {
  __shared__ float s_esc[32][10];  // 1.25 KB
  __shared__ float s_h[32][64];    // 8 KB
  __shared__ float s_w[32][192];   // 24 KB
  const int tid = threadIdx.x;
  const int e0 = blockIdx.x * 32;

  // Phase 1: stage esc tile
  for (int idx = tid; idx < 320; idx += 256)
    s_esc[idx / 10][idx % 10] = esc[(size_t)(e0 + idx / 10) * 10 + idx % 10];
  __syncthreads();

  // Phase 2: fc1 + silu (K=10): thread -> (edge i, 8 outputs)
  {
    const int i = tid >> 3;
    const int o0 = (tid & 7) * 8;
    const float sc = 0.31622776601683794f;  // 1/sqrt(10)
    for (int o = o0; o < o0 + 8; ++o) {
      float s = 0.f;
#pragma unroll
      for (int k = 0; k < 10; ++k) s += s_esc[i][k] * Wfc1[k * 64 + o];
      s *= sc;
      s_h[i][o] = s / (1.f + __expf(-s));  // silu
    }
  }
  __syncthreads();

  // Phase 3: fc2 via WMMA: (32x64)@(64x192), 2x12=24 tiles over 8 waves
  {
    const int wave = tid >> 5, lane = tid & 31;
    for (int t = wave; t < 24; t += 8) {
      const int m0 = (t / 12) * 16, n0 = (t % 12) * 16;
      v8f acc = gemm_tile<1>(&s_h[0][0], 64, Wfc2, 192, m0, n0, 64, lane);
      store_tile<false, false>(acc, &s_w[0][0], 192, 1, m0, n0, lane, 0.125f,
                               nullptr, nullptr, 0, 0);
    }
  }
  __syncthreads();

  // Phase 4: messages + scatter. 8 threads per edge.
  {
    const int el = tid >> 3, sub = tid & 7;
    const int e = e0 + el;
    const int s = src[e], d = dst[e];
    const float sh0 = sh[(size_t)e * 4 + 0];
    const float sv0 = sh[(size_t)e * 4 + 1];
    const float sv1 = sh[(size_t)e * 4 + 2];
    const float sv2 = sh[(size_t)e * 4 + 3];
    const float invsqrt3 = 0.5773502691896258f;
    // scalar channels [sub*8, sub*8+8)
#pragma unroll
    for (int c = 0; c < 8; ++c) {
      const int ch = sub * 8 + c;
      const float g = hs[(size_t)s * 64 + ch];
      const float w1 = s_w[el][ch];
      const float w2 = s_w[el][64 + ch];
      unsafeAtomicAdd(&Ms[(size_t)d * 96 + ch], g * sh0 * w1);
      const float gw = g * w2;
      const size_t vb = ((size_t)d * 96 + ch) * 3;
      unsafeAtomicAdd(&Mv[vb + 0], gw * sv0);
      unsafeAtomicAdd(&Mv[vb + 1], gw * sv1);
      unsafeAtomicAdd(&Mv[vb + 2], gw * sv2);
    }
    // vector channels [sub*4, sub*4+4)
#pragma unroll
    for (int c = 0; c < 4; ++c) {
      const int cv = sub * 4 + c;
      const size_t gb = ((size_t)s * 32 + cv) * 3;
      const float gx = hv[gb], gy = hv[gb + 1], gz = hv[gb + 2];
      const float w3 = s_w[el][128 + cv];
      const float w4 = s_w[el][160 + cv];
      const float dot = gx * sv0 + gy * sv1 + gz * sv2;
      unsafeAtomicAdd(&Ms[(size_t)d * 96 + 64 + cv], dot * invsqrt3 * w4);
      const float f = sh0 * w3;
      const size_t vb = ((size_t)d * 96 + 64 + cv) * 3;
      unsafeAtomicAdd(&Mv[vb + 0], gx * f);
      unsafeAtomicAdd(&Mv[vb + 1], gy * f);
      unsafeAtomicAdd(&Mv[vb + 2], gz * f);
    }
    if (sub == 0) unsafeAtomicAdd(&deg[d], 1.0f);
  }
}

// Normalize accumulators by degree (clamped to >=1).
__global__ void norm_kernel(float* Ms, float* Mv, const float* __restrict__ deg, int n) {
  const int t = blockIdx.x * blockDim.x + threadIdx.x;
  if (t >= n * 96) return;
  const int node = t / 96;
  const float inv = 1.f / fmaxf(deg[node], 1.f);
  Ms[t] *= inv;
  const size_t vb = (size_t)t * 3;
  Mv[vb + 0] *= inv;
  Mv[vb + 1] *= inv;
  Mv[vb + 2] *= inv;
}

// alphaAttr[n] = (Ms[n,:] . Wa) / sqrt(96) * attr[n]^2
// One wave32 per node: coalesced lane-strided reads + shfl_xor butterfly.
__global__ __launch_bounds__(256) void alpha_kernel(
    const float* __restrict__ Ms, const float* __restrict__ Wa,
    const float* __restrict__ attr, float* alphaAttr, int n) {
  const int node = (blockIdx.x * 256 + threadIdx.x) >> 5;
  const int lane = threadIdx.x & 31;
  if (node >= n) return;
  float a = 0.f;
#pragma unroll
  for (int k = lane; k < 96; k += 32) a += Ms[(size_t)node * 96 + k] * Wa[k];
#pragma unroll
  for (int off = 16; off > 0; off >>= 1) a += __shfl_xor(a, off, 32);
  if (lane == 0) {
    const float at = attr[node];
    alphaAttr[node] = a * 0.10206207261596575f * at * at;  // 1/sqrt(96)
  }
}

// Final per-layer combine: out_s = sc_s + alphaAttr * (Ms @ Wl2_s / sqrt(96))
//                          out_v = sc_v + alphaAttr * (Mv @ Wl2_v / sqrt(96))
// Layer 2 writes straight into the concatenated d_out via ld=160 strides.
__global__ __launch_bounds__(128) void node_out_kernel(
    const float* __restrict__ Ms, const float* __restrict__ Mv,
    const float* __restrict__ Wl2_s, const float* __restrict__ Wl2_v,
    const float* __restrict__ sc_s, const float* __restrict__ sc_v,
    const float* __restrict__ alphaAttr, float* out_s, int ldOutS,
    float* out_v, int ldOutV, int o_s, int nNodes) {
  const int lane = threadIdx.x & 31;
  const int m0 = blockIdx.x * 64 + (threadIdx.x >> 5) * 16;
  if (m0 >= nNodes) return;
  const float sc = 0.10206207261596575f;  // 1/sqrt(96)
  for (int nt = 0; nt < o_s; nt += 16) {
    v8f acc = gemm_tile<1>(Ms, 96, Wl2_s, o_s, m0, nt, 96, lane);
    store_tile<true, true>(acc, out_s, ldOutS, 1, m0, nt, lane, sc, alphaAttr,
                           sc_s, o_s, 1);
  }
#pragma unroll
  for (int i = 0; i < 3; ++i) {
    for (int nt = 0; nt < 32; nt += 16) {
      v8f acc = gemm_tile<3>(Mv + i, 288, Wl2_v, 32, m0, nt, 96, lane);
      store_tile<true, true>(acc, out_v + i, ldOutV, 3, m0, nt, lane, sc,
                             alphaAttr, sc_v + i, 96, 3);
    }
  }
}

// Gate between layers: gs = silu(s1[:, :64]); gv = v1 * sigmoid(s1[:, 64:96])
__global__ void gate_kernel(const float* __restrict__ s1, const float* __restrict__ v1,
                            float* gs, float* gv, int n) {
  const int t = blockIdx.x * blockDim.x + threadIdx.x;
  if (t >= n * 64) return;
  const int node = t >> 6, ch = t & 63;
  const float x = s1[(size_t)node * 96 + ch];
  gs[t] = x / (1.f + __expf(-x));
  if (ch < 32) {
    const float y = s1[(size_t)node * 96 + 64 + ch];
    const float f = 1.f / (1.f + __expf(-y));
    const size_t vb = ((size_t)node * 32 + ch) * 3;
    gv[vb + 0] = v1[vb + 0] * f;
    gv[vb + 1] = v1[vb + 1] * f;
    gv[vb + 2] = v1[vb + 2] * f;
  }
}

extern "C" void kernel_launch(void* const* d_in, const int* in_sizes, int n_in,
                              void* d_out, int out_size, void* d_ws, size_t ws_size,
                              hipStream_t stream) {
  (void)in_sizes; (void)n_in; (void)out_size; (void)ws_size;
  const int N = N_NODES, E = N_EDGES;
  const float* node_s = (const float*)d_in[0];
  const float* node_v = (const float*)d_in[1];
  const float* attr   = (const float*)d_in[2];
  const int*   esrc   = (const int*)d_in[3];
  const int*   edst   = (const int*)d_in[4];
  const float* esh    = (const float*)d_in[5];
  const float* escal  = (const float*)d_in[6];
  // per-layer weights: [sc_s, sc_v, lin1_s, lin1_v, fc1, fc2, lin2_s, lin2_v, alpha]
  const float* const* L1 = (const float* const*)&d_in[7];
  const float* const* L2 = (const float* const*)&d_in[16];

  float* ws = (float*)d_ws;
  float* hs = ws;        ws += (size_t)N * 64;
  float* hv = ws;        ws += (size_t)N * 96;
  float* sc_s = ws;      ws += (size_t)N * 96;
  float* sc_v = ws;      ws += (size_t)N * 96;
  float* Ms = ws;        ws += (size_t)N * 96;   // Ms, Mv, deg contiguous
  float* Mv = ws;        ws += (size_t)N * 288;  //   -> single memset
  float* deg = ws;       ws += (size_t)N;
  float* alphaAttr = ws; ws += (size_t)N;
  float* s1 = ws;        ws += (size_t)N * 96;
  float* v1 = ws;        ws += (size_t)N * 96;
  float* gs = ws;        ws += (size_t)N * 64;
  float* gv = ws;        ws += (size_t)N * 96;

  const int NBLK_NODE = (N + 63) / 64;
  const size_t accBytes = (size_t)N * (96 + 288 + 1) * sizeof(float);

  // -------- Layer 1 (o_s = 96) --------
  node_linear_kernel<<<NBLK_NODE, 128, 0, stream>>>(
      node_s, node_v, attr, L1[0], L1[1], L1[2], L1[3], hs, hv, sc_s, sc_v, 96, N);
  hipMemsetAsync(Ms, 0, accBytes, stream);
  edge_kernel<<<E / 32, 256, 0, stream>>>(escal, L1[4], L1[5], hs, hv, esrc, edst,
                                          esh, Ms, Mv, deg);
  norm_kernel<<<(N * 96 + 255) / 256, 256, 0, stream>>>(Ms, Mv, deg, N);
  alpha_kernel<<<(N * 32 + 255) / 256, 256, 0, stream>>>(Ms, L1[8], attr, alphaAttr, N);
  node_out_kernel<<<NBLK_NODE, 128, 0, stream>>>(Ms, Mv, L1[6], L1[7], sc_s, sc_v,
                                                 alphaAttr, s1, 96, v1, 96, 96, N);
  gate_kernel<<<(N * 64 + 255) / 256, 256, 0, stream>>>(s1, v1, gs, gv, N);

  // -------- Layer 2 (o_s = 64), writes into concatenated d_out (N,160) --------
  node_linear_kernel<<<NBLK_NODE, 128, 0, stream>>>(
      gs, gv, attr, L2[0], L2[1], L2[2], L2[3], hs, hv, sc_s, sc_v, 64, N);
  hipMemsetAsync(Ms, 0, accBytes, stream);
  edge_kernel<<<E / 32, 256, 0, stream>>>(escal, L2[4], L2[5], hs, hv, esrc, edst,
                                          esh, Ms, Mv, deg);
  norm_kernel<<<(N * 96 + 255) / 256, 256, 0, stream>>>(Ms, Mv, deg, N);
  alpha_kernel<<<(N * 32 + 255) / 256, 256, 0, stream>>>(Ms, L2[8], attr, alphaAttr, N);
  float* out = (float*)d_out;
  node_out_kernel<<<NBLK_NODE, 128, 0, stream>>>(Ms, Mv, L2[6], L2[7], sc_s, sc_v,
                                                 alphaAttr, out, 160, out + 64, 160,
                                                 64, N);
}